// SimulTransOracle_74680891343571
// MI455X (gfx1250) — compile-verified
//
#include <hip/hip_runtime.h>
#include <float.h>

#define NB    32      // batches
#define TT    1024    // target length
#define TS    1024    // source length
#define NT    256     // threads per block
#define NW    8       // waves per block (wave32)
#define VEC   4       // columns per thread
#define DEPTH 8       // async pipeline depth (rows in flight)

#define HAS_ASYNC_LDS (__has_builtin(__builtin_amdgcn_global_load_async_to_lds_b128))

#if __has_builtin(__builtin_amdgcn_s_wait_asynccnt)
#define WAIT_ASYNC(n) do { __builtin_amdgcn_s_wait_asynccnt(n); asm volatile("" ::: "memory"); } while (0)
#else
#define WAIT_ASYNC(n) asm volatile("s_wait_asynccnt %0" :: "i"(n) : "memory")
#endif

// LDS reads must drain before an async DMA may overwrite the buffer
#define WAIT_DS0() asm volatile("s_wait_dscnt 0x0" ::: "memory")

#if HAS_ASYNC_LDS
typedef int v4i_t __attribute__((ext_vector_type(4)));
typedef __attribute__((address_space(1))) v4i_t* gptr128_t;   // global
typedef __attribute__((address_space(3))) v4i_t* lptr128_t;   // LDS

__device__ __forceinline__ void async_copy16(const float* g, float* l) {
  // global (AS1) -> LDS (AS3), 16 bytes per lane, tracked by ASYNCcnt
  __builtin_amdgcn_global_load_async_to_lds_b128(
      (gptr128_t)(g), (lptr128_t)(l), 0, 0);
}

__device__ __forceinline__ void wait_async_le(int n) {
  switch (n) {
    case 0: WAIT_ASYNC(0); break;
    case 1: WAIT_ASYNC(1); break;
    case 2: WAIT_ASYNC(2); break;
    case 3: WAIT_ASYNC(3); break;
    case 4: WAIT_ASYNC(4); break;
    case 5: WAIT_ASYNC(5); break;
    case 6: WAIT_ASYNC(6); break;
    default: WAIT_ASYNC(7); break;
  }
}
#endif

__global__ __launch_bounds__(NT) void simul_trans_oracle_kernel(
    const float* __restrict__ scores,   // (NB, TT, TS) f32
    int* __restrict__ out,              // (NB, TT) i32
    float* __restrict__ ws_cs)          // (NB, TT+1, TS) f32 scratch
{
  const int b    = blockIdx.x;
  const int tid  = threadIdx.x;
  const int lane = tid & 31;
  const int wv   = tid >> 5;
  const int j0   = tid * VEC;

  const float* sc = scores + (size_t)b * TT * TS;
  float* fs = ws_cs + (size_t)b * (TT + 1) * TS;   // holds fs, then cs in place
  int* outb = out + b * TT;

#if HAS_ASYNC_LDS
  __shared__ float sbuf[DEPTH][TS];
#endif
  __shared__ float c0sh[TT];
  __shared__ float waveAgg[2][NW];
  __shared__ float tile[65][66];
  __shared__ int   bestSh[TT];
  __shared__ int   tjs[2];

  const float INFV = __builtin_inff();

  // Exact dyadic-rational DP constants (bit-exact vs cumsum of r[j]=(j+1)/1024)
  float Rv[VEC], Sv[VEC], row0[VEC];
  #pragma unroll
  for (int k = 0; k < VEC; ++k) {
    const long long j = j0 + k;
    Rv[k]   = (float)((double)((j + 1) * (j + 2) / 2 - 1) / 1024.0); // R[j]
    Sv[k]   = (float)((double)(j * (j + 1) / 2) / 1024.0);           // S[j]
    row0[k] = (float)((double)((j + 1) * (j + 2) / 2) / 1024.0);     // cumsum(r)[j]
  }

  // ---------------- forward DP:  fs[t+1] = R + cummin(a - R) ----------------
  float prev[VEC];
  #pragma unroll
  for (int k = 0; k < VEC; ++k) prev[k] = row0[k];
  *(float4*)(fs + j0) = make_float4(prev[0], prev[1], prev[2], prev[3]);

  float c0run = 0.0f;

#if HAS_ASYNC_LDS
  #pragma unroll
  for (int p = 0; p < DEPTH; ++p)
    async_copy16(sc + (size_t)p * TS + j0, &sbuf[p][j0]);
#endif
  for (int t = 0; t < TT; ++t) {
    const int buf = t & 1;
    float s0, s1, s2, s3;
#if HAS_ASYNC_LDS
    const int ib = t & (DEPTH - 1);
    wait_async_le(min(DEPTH - 1, TT - 1 - t));      // row t has landed in LDS
    s0 = -sbuf[ib][j0 + 0]; s1 = -sbuf[ib][j0 + 1];
    s2 = -sbuf[ib][j0 + 2]; s3 = -sbuf[ib][j0 + 3];
#else
    const float4 sv = *(const float4*)(sc + (size_t)t * TS + j0);
    if (t + 1 < TT) __builtin_prefetch(sc + (size_t)(t + 1) * TS + j0, 0, 0);
    s0 = -sv.x; s1 = -sv.y; s2 = -sv.z; s3 = -sv.w;
#endif
    float a0 = prev[0] + s0, a1 = prev[1] + s1, a2 = prev[2] + s2, a3 = prev[3] + s3;
    if (tid == 0) { c0run += s0; a0 = c0run; c0sh[t] = c0run; }
    const float m0 = a0 - Rv[0], m1 = a1 - Rv[1], m2 = a2 - Rv[2], m3 = a3 - Rv[3];
    // local inclusive prefix-min over the thread's 4 columns
    const float l0 = m0;
    const float l1 = fminf(l0, m1);
    const float l2 = fminf(l1, m2);
    const float l3 = fminf(l2, m3);
    // wave-level inclusive scan of per-thread totals
    float tp = l3;
    #pragma unroll
    for (int d = 1; d < 32; d <<= 1) {
      const float o = __shfl_up(tp, d, 32);
      if (lane >= d) tp = fminf(tp, o);
    }
    if (lane == 31) waveAgg[buf][wv] = tp;
    float texcl = __shfl_up(tp, 1, 32);
    if (lane == 0) texcl = INFV;
    __syncthreads();
    float off = texcl;
    #pragma unroll
    for (int w = 0; w < NW - 1; ++w) {
      const float v = waveAgg[buf][w];
      off = (w < wv) ? fminf(off, v) : off;
    }
    const float f0 = Rv[0] + fminf(l0, off);
    const float f1 = Rv[1] + fminf(l1, off);
    const float f2 = Rv[2] + fminf(l2, off);
    const float f3 = Rv[3] + fminf(l3, off);
    prev[0] = f0; prev[1] = f1; prev[2] = f2; prev[3] = f3;
    *(float4*)(fs + (size_t)(t + 1) * TS + j0) = make_float4(f0, f1, f2, f3);
#if HAS_ASYNC_LDS
    if (t + DEPTH < TT) {               // refill the buffer just consumed
      WAIT_DS0();                       // prior ds reads must drain first
      async_copy16(sc + (size_t)(t + DEPTH) * TS + j0, &sbuf[ib][j0]);
    }
#endif
  }

  __syncthreads();

  // ---------------- backward DP fused with cs = fs + bs ----------------
  float prevb[VEC];
  #pragma unroll
  for (int k = 0; k < VEC; ++k)
    prevb[k] = (float)((double)(TS - (j0 + k)) / 1024.0);   // bs_last[j] = r[Ts-1-j]
  {
    const float4 fv = *(const float4*)(fs + (size_t)TT * TS + j0);
    *(float4*)(fs + (size_t)TT * TS + j0) =
        make_float4(fv.x + prevb[0], fv.y + prevb[1], fv.z + prevb[2], fv.w + prevb[3]);
  }

#if HAS_ASYNC_LDS
  #pragma unroll
  for (int p = 0; p < DEPTH; ++p)
    async_copy16(sc + (size_t)(TT - 1 - p) * TS + j0, &sbuf[p][j0]);
#endif
  for (int t = TT - 1; t >= 0; --t) {
    const int buf = (TT - 1 - t) & 1;
    float s0, s1, s2, s3;
#if HAS_ASYNC_LDS
    const int ib = (TT - 1 - t) & (DEPTH - 1);
    wait_async_le(min(DEPTH - 1, t));               // row t has landed in LDS
    s0 = -sbuf[ib][j0 + 0]; s1 = -sbuf[ib][j0 + 1];
    s2 = -sbuf[ib][j0 + 2]; s3 = -sbuf[ib][j0 + 3];
#else
    const float4 sv = *(const float4*)(sc + (size_t)t * TS + j0);
    if (t > 0) __builtin_prefetch(sc + (size_t)(t - 1) * TS + j0, 0, 0);
    s0 = -sv.x; s1 = -sv.y; s2 = -sv.z; s3 = -sv.w;
#endif
    if (t >= DEPTH) __builtin_prefetch(fs + (size_t)(t - DEPTH) * TS + j0, 0, 0);
    const float4 fv = *(const float4*)(fs + (size_t)t * TS + j0);  // fs row (early issue)
    const float c0v = c0sh[TT - 1 - t];
    float w0 = prevb[0] + s0, w1 = prevb[1] + s1, w2 = prevb[2] + s2, w3 = prevb[3] + s3;
    if (tid == NT - 1) w3 = c0v;                       // column Ts-1 preset
    const float m0 = w0 + Sv[0], m1 = w1 + Sv[1], m2 = w2 + Sv[2], m3 = w3 + Sv[3];
    // local inclusive suffix-min
    const float l3 = m3;
    const float l2 = fminf(m2, l3);
    const float l1 = fminf(m1, l2);
    const float l0 = fminf(m0, l1);
    float tp = l0;
    #pragma unroll
    for (int d = 1; d < 32; d <<= 1) {
      const float o = __shfl_down(tp, d, 32);
      if (lane < 32 - d) tp = fminf(tp, o);
    }
    if (lane == 0) waveAgg[buf][wv] = tp;
    float texcl = __shfl_down(tp, 1, 32);
    if (lane == 31) texcl = INFV;
    __syncthreads();
    float off = texcl;
    #pragma unroll
    for (int w = 1; w < NW; ++w) {
      const float v = waveAgg[buf][w];
      off = (w > wv) ? fminf(off, v) : off;
    }
    const float b0 = fminf(l0, off) - Sv[0];
    const float b1 = fminf(l1, off) - Sv[1];
    const float b2 = fminf(l2, off) - Sv[2];
    const float b3 = fminf(l3, off) - Sv[3];
    prevb[0] = b0; prevb[1] = b1; prevb[2] = b2; prevb[3] = b3;
    *(float4*)(fs + (size_t)t * TS + j0) =
        make_float4(fv.x + b0, fv.y + b1, fv.z + b2, fv.w + b3);   // cs row
#if HAS_ASYNC_LDS
    if (t - DEPTH >= 0) {               // refill the buffer just consumed
      WAIT_DS0();
      async_copy16(sc + (size_t)(t - DEPTH) * TS + j0, &sbuf[ib][j0]);
    }
#endif
  }

  __threadfence();   // make this block's cs stores visible to all its waves
  __syncthreads();

  // ---------------- greedy path walk, tiled through LDS ----------------
  for (int i = tid; i < TT; i += NT) bestSh[i] = TS - 1;

  int t = 0, jj = 0;
  while (t < TT && jj < TS - 1) {
    const int t0  = t & ~63;
    const int jc0 = jj & ~63;
    const int nc  = min(65, TS - jc0);
    __syncthreads();
    for (int idx = tid; idx < 65 * 65; idx += NT) {
      const int rr = idx / 65, cc = idx % 65;
      float v = FLT_MAX;
      if (cc < nc) v = fs[(size_t)(t0 + rr) * TS + (jc0 + cc)];
      tile[rr][cc] = v;
    }
    __syncthreads();
    if (tid == 0) {
      int tt = t, jl = jj;
      while (tt < TT && jl < TS - 1 && tt < t0 + 64 && jl < jc0 + 64) {
        const float cw = tile[tt + 1 - t0][jl - jc0];
        const float cr = tile[tt - t0][jl + 1 - jc0];
        if (cw < cr) { bestSh[tt] = jl; ++tt; }
        else         { ++jl; }
      }
      tjs[0] = tt; tjs[1] = jl;
    }
    __syncthreads();
    t = tjs[0]; jj = tjs[1];
  }

  __syncthreads();
  for (int i = tid; i < TT; i += NT) outb[i] = bestSh[i];
}

extern "C" void kernel_launch(void* const* d_in, const int* in_sizes, int n_in,
                              void* d_out, int out_size, void* d_ws, size_t ws_size,
                              hipStream_t stream) {
  (void)in_sizes; (void)n_in; (void)out_size; (void)ws_size;
  const float* scores = (const float*)d_in[0];
  int* out  = (int*)d_out;
  float* cs = (float*)d_ws;   // needs NB*(TT+1)*TS*4 ≈ 134 MB
  simul_trans_oracle_kernel<<<dim3(NB), dim3(NT), 0, stream>>>(scores, out, cs);
}